// HopfActCpx_39934605918624
// MI455X (gfx1250) — compile-verified
//
#include <hip/hip_runtime.h>

// HopfActCpx: dz/dt = (a + b|z|^2) z, fixed-step Dormand-Prince 5(4),
// 256 steps over [0, 2*pi]. 16.7M independent ODEs, f32.
// Compute-bound (~3.8e11 fused f32 ops vs 536 MB traffic) -> target the
// gfx1250 packed-FP32 VALU path (v_pk_fma_f32) with <2 x float> vectors.

typedef __attribute__((ext_vector_type(2))) float f32x2;
typedef __attribute__((ext_vector_type(4))) float f32x4;

namespace {

constexpr double DT = 6.283185307179586476925286766559 / 256.0; // 2*pi/256

// dt-folded Dormand-Prince 5(4) tableau (5th-order propagating solution)
constexpr float C21 = (float)(DT * (1.0 / 5.0));
constexpr float C31 = (float)(DT * (3.0 / 40.0));
constexpr float C32 = (float)(DT * (9.0 / 40.0));
constexpr float C41 = (float)(DT * (44.0 / 45.0));
constexpr float C42 = (float)(DT * (-56.0 / 15.0));
constexpr float C43 = (float)(DT * (32.0 / 9.0));
constexpr float C51 = (float)(DT * (19372.0 / 6561.0));
constexpr float C52 = (float)(DT * (-25360.0 / 2187.0));
constexpr float C53 = (float)(DT * (64448.0 / 6561.0));
constexpr float C54 = (float)(DT * (-212.0 / 729.0));
constexpr float C61 = (float)(DT * (9017.0 / 3168.0));
constexpr float C62 = (float)(DT * (-355.0 / 33.0));
constexpr float C63 = (float)(DT * (46732.0 / 5247.0));
constexpr float C64 = (float)(DT * (49.0 / 176.0));
constexpr float C65 = (float)(DT * (-5103.0 / 18656.0));
constexpr float D1  = (float)(DT * (35.0 / 384.0));
constexpr float D3  = (float)(DT * (500.0 / 1113.0));
constexpr float D4  = (float)(DT * (125.0 / 192.0));
constexpr float D5  = (float)(DT * (-2187.0 / 6784.0));
constexpr float D6  = (float)(DT * (11.0 / 84.0));

__device__ __forceinline__ f32x2 splat(float s) {
    f32x2 v; v.x = s; v.y = s; return v;
}

// packed fused multiply-add on <2 x float> -> v_pk_fma_f32 on gfx1250
__device__ __forceinline__ f32x2 vfma(f32x2 a, f32x2 b, f32x2 c) {
    return __builtin_elementwise_fma(a, b, c);
}
__device__ __forceinline__ f32x2 cfma(float a, f32x2 b, f32x2 c) {
    return __builtin_elementwise_fma(splat(a), b, c);
}

struct Params { f32x2 ar, ai, br, bi; };

// f(z) = (a + b|z|^2) z  in real arithmetic; 8 fused packed ops.
__device__ __forceinline__ void hopf_rhs(const Params& p, f32x2 x, f32x2 y,
                                         f32x2& fx, f32x2& fy) {
    f32x2 r2 = vfma(y, y, x * x);         // |z|^2
    f32x2 wr = vfma(p.br, r2, p.ar);      // Re(a + b r2)
    f32x2 wi = vfma(p.bi, r2, p.ai);      // Im(a + b r2)
    fx = vfma(wr, x, -(wi * y));          // neg is a free src modifier
    fy = vfma(wr, y, wi * x);
}

} // namespace

__global__ __launch_bounds__(256)
void hopf_dopri5_kernel(const float* __restrict__ z_re,
                        const float* __restrict__ z_im,
                        const float* __restrict__ a_re,
                        const float* __restrict__ a_im,
                        const float* __restrict__ b_re,
                        const float* __restrict__ b_im,
                        float* __restrict__ out, long long n) {
    long long base = ((long long)blockIdx.x * blockDim.x + threadIdx.x) * 4;
    if (base >= n) return;  // n is a multiple of 4 (4096*4096)

    // 128-bit coalesced loads: 4 elements per lane = two f32x2 packs (ILP).
    f32x4 vzr = *(const f32x4*)(z_re + base);
    f32x4 vzi = *(const f32x4*)(z_im + base);
    f32x4 var = *(const f32x4*)(a_re + base);
    f32x4 vai = *(const f32x4*)(a_im + base);
    f32x4 vbr = *(const f32x4*)(b_re + base);
    f32x4 vbi = *(const f32x4*)(b_im + base);

    f32x2 x[2], y[2];
    Params p[2];
    x[0] = {vzr.x, vzr.y};  x[1] = {vzr.z, vzr.w};
    y[0] = {vzi.x, vzi.y};  y[1] = {vzi.z, vzi.w};
    p[0].ar = {var.x, var.y};  p[1].ar = {var.z, var.w};
    p[0].ai = {vai.x, vai.y};  p[1].ai = {vai.z, vai.w};
    p[0].br = {vbr.x, vbr.y};  p[1].br = {vbr.z, vbr.w};
    p[0].bi = {vbi.x, vbi.y};  p[1].bi = {vbi.z, vbi.w};

#pragma unroll 1
    for (int step = 0; step < 256; ++step) {
#pragma unroll
        for (int q = 0; q < 2; ++q) {
            f32x2 k1x, k1y, k2x, k2y, k3x, k3y, k4x, k4y, k5x, k5y, k6x, k6y;
            f32x2 sx, sy;

            hopf_rhs(p[q], x[q], y[q], k1x, k1y);

            sx = cfma(C21, k1x, x[q]);
            sy = cfma(C21, k1y, y[q]);
            hopf_rhs(p[q], sx, sy, k2x, k2y);

            sx = cfma(C31, k1x, cfma(C32, k2x, x[q]));
            sy = cfma(C31, k1y, cfma(C32, k2y, y[q]));
            hopf_rhs(p[q], sx, sy, k3x, k3y);

            sx = cfma(C41, k1x, cfma(C42, k2x, cfma(C43, k3x, x[q])));
            sy = cfma(C41, k1y, cfma(C42, k2y, cfma(C43, k3y, y[q])));
            hopf_rhs(p[q], sx, sy, k4x, k4y);

            sx = cfma(C51, k1x, cfma(C52, k2x, cfma(C53, k3x, cfma(C54, k4x, x[q]))));
            sy = cfma(C51, k1y, cfma(C52, k2y, cfma(C53, k3y, cfma(C54, k4y, y[q]))));
            hopf_rhs(p[q], sx, sy, k5x, k5y);

            sx = cfma(C61, k1x, cfma(C62, k2x, cfma(C63, k3x,
                 cfma(C64, k4x, cfma(C65, k5x, x[q])))));
            sy = cfma(C61, k1y, cfma(C62, k2y, cfma(C63, k3y,
                 cfma(C64, k4y, cfma(C65, k5y, y[q])))));
            hopf_rhs(p[q], sx, sy, k6x, k6y);

            x[q] = cfma(D1, k1x, cfma(D3, k3x, cfma(D4, k4x,
                   cfma(D5, k5x, cfma(D6, k6x, x[q])))));
            y[q] = cfma(D1, k1y, cfma(D3, k3y, cfma(D4, k4y,
                   cfma(D5, k5y, cfma(D6, k6y, y[q])))));
        }
    }

    f32x4 ore = {x[0].x, x[0].y, x[1].x, x[1].y};
    f32x4 oim = {y[0].x, y[0].y, y[1].x, y[1].y};
    *(f32x4*)(out + base)     = ore;   // out[0, :, :] = Re z(T)
    *(f32x4*)(out + n + base) = oim;   // out[1, :, :] = Im z(T)
}

extern "C" void kernel_launch(void* const* d_in, const int* in_sizes, int n_in,
                              void* d_out, int out_size, void* d_ws, size_t ws_size,
                              hipStream_t stream) {
    (void)n_in; (void)out_size; (void)d_ws; (void)ws_size;
    const float* z_re = (const float*)d_in[0];
    const float* z_im = (const float*)d_in[1];
    const float* a_re = (const float*)d_in[2];
    const float* a_im = (const float*)d_in[3];
    const float* b_re = (const float*)d_in[4];
    const float* b_im = (const float*)d_in[5];
    float* out = (float*)d_out;

    long long n = (long long)in_sizes[0];          // 4096*4096 elements
    long long threads = (n + 3) / 4;               // 4 elements per lane
    const int block = 256;                         // 8 wave32s
    long long grid = (threads + block - 1) / block;

    hopf_dopri5_kernel<<<dim3((unsigned)grid), dim3(block), 0, stream>>>(
        z_re, z_im, a_re, a_im, b_re, b_im, out, n);
}